// RecentCopyBias_13486197310065
// MI455X (gfx1250) — compile-verified
//
#include <hip/hip_runtime.h>
#include <hip/hip_bf16.h>
#include <math.h>

// Problem constants (fixed by the reference).
#define BB 4
#define TT 1024
#define DD 1024
#define VV 32000
#define WW 8
#define ROWS (BB * TT)          // 4096 (b*t rows)
#define Q_PER_ROW (VV / 4)      // 8000 float4 per vocab row

typedef float v2f __attribute__((ext_vector_type(2)));
typedef float v4f __attribute__((ext_vector_type(4)));
typedef float v8f __attribute__((ext_vector_type(8)));

// ---------------------------------------------------------------------------
// Kernel A: gate[r] = sigmoid(hidden[r,:] . gate_w + gate_b) via
// V_WMMA_F32_16X16X4_F32, plus lw[j] = copy_scale * softmax(lag_logits)[j].
//
// One wave handles 16 rows (positions). A-matrix layout (ISA 7.12.2, 32-bit
// A 16x4): lane L holds row M = L%16; VGPR0/1 hold K = kb + 2*(L>=16) + {0,1}.
// B-matrix (4x16) mirrors A with M<->N, and we replicate gate_w across all N,
// so every lane loads the same float2 of weights for its K pair. Every column
// of D then carries the same per-row dot product; lanes with N==0 write out.
// ---------------------------------------------------------------------------
__global__ void gate_wmma_kernel(const float* __restrict__ hidden,
                                 const float* __restrict__ gate_w,
                                 const float* __restrict__ gate_b,
                                 const float* __restrict__ lag_logits,
                                 const float* __restrict__ copy_scale,
                                 float* __restrict__ gate_out,
                                 float* __restrict__ lw_out) {
    const int lane  = threadIdx.x & 31;
    const int wave  = (blockIdx.x * blockDim.x + threadIdx.x) >> 5;
    const int pos0  = wave * 16;              // 16 rows per wave
    const int m     = lane & 15;              // A row this lane covers
    const int khalf = lane >> 4;              // 0: K=kb+{0,1}, 1: K=kb+{2,3}

    const float* rowp = hidden + (size_t)(pos0 + m) * DD;

    v8f c = {0.f, 0.f, 0.f, 0.f, 0.f, 0.f, 0.f, 0.f};
    #pragma unroll 4
    for (int kb = 0; kb < DD; kb += 4) {
        const int k = kb + khalf * 2;
        v2f a  = *(const v2f*)(rowp + k);     // hidden[row][k], hidden[row][k+1]
        v2f bv = *(const v2f*)(gate_w + k);   // w[k], w[k+1] replicated over N
        c = __builtin_amdgcn_wmma_f32_16x16x4_f32(
            /*neg_a=*/false, a, /*neg_b=*/false, bv,
            /*c_mod=*/(short)0, c, /*reuse_a=*/false, /*reuse_b=*/false);
    }

    // D layout: lane L holds (M = 8*(L>=16)+v, N = L%16). Column N==0 lanes
    // (lanes 0 and 16) jointly hold all 16 row results.
    if ((lane & 15) == 0) {
        const float gb   = gate_b[0];
        const int mbase  = khalf * 8;
        #pragma unroll
        for (int i = 0; i < 8; ++i) {
            const float x = c[i] + gb;
            gate_out[pos0 + mbase + i] = 1.0f / (1.0f + __expf(-x));
        }
    }

    // Tiny softmax of 8 lag logits, folded with copy_scale, done once.
    if (blockIdx.x == 0 && threadIdx.x == 0) {
        const float cs = copy_scale[0];
        float ll[WW];
        float mx = -INFINITY;
        #pragma unroll
        for (int j = 0; j < WW; ++j) { ll[j] = lag_logits[j]; mx = fmaxf(mx, ll[j]); }
        float sum = 0.f;
        #pragma unroll
        for (int j = 0; j < WW; ++j) { ll[j] = __expf(ll[j] - mx); sum += ll[j]; }
        const float inv = cs / sum;
        #pragma unroll
        for (int j = 0; j < WW; ++j) lw_out[j] = ll[j] * inv;
    }
}

// ---------------------------------------------------------------------------
// Kernel B: one block per (b,t) row. Threads 0..7 compute the 8 scatter
// tuples into LDS; each thread then pre-filters to the float4 slots it owns
// (owner thread = slot % 256) into fully-unrolled register arrays, and
// streams the 32000-float row with non-temporal b128 stores, merging owned
// contributions in-register. No atomics, no cross-block conflicts.
// ---------------------------------------------------------------------------
__global__ void __launch_bounds__(256)
fill_scatter_kernel(const int* __restrict__ input_ids,
                    const float* __restrict__ gate,
                    const float* __restrict__ lw,
                    float* __restrict__ out) {
    const int tid = threadIdx.x;
    const int r   = blockIdx.x;          // flat (b,t) row
    const int t   = r & (TT - 1);
    const int b   = r >> 10;

    __shared__ int   s_q[WW];            // float4 slot index (tok >> 2), -1 invalid
    __shared__ int   s_comp[WW];         // component (tok & 3)
    __shared__ float s_val[WW];          // contribution value

    if (tid < WW) {
        const int lag = tid;
        int q = -1, comp = 0;
        float val = 0.f;
        if (t >= lag) {
            const int tok = input_ids[b * TT + t - lag];
            val  = gate[r] * lw[lag];
            q    = tok >> 2;
            comp = tok & 3;
        }
        s_q[tid] = q; s_comp[tid] = comp; s_val[tid] = val;
    }
    __syncthreads();

    // Ownership pre-filter into registers (static indexing, no scratch).
    int      m_it[WW];
    int      m_comp[WW];
    float    m_val[WW];
    unsigned own_mask = 0u;
    #pragma unroll
    for (int j = 0; j < WW; ++j) {
        const int q = s_q[j];
        const bool own = (q >= 0) && ((q & 255) == tid);
        m_it[j]   = own ? (q >> 8) : -1;
        m_comp[j] = s_comp[j];
        m_val[j]  = s_val[j];
        own_mask |= own ? (1u << j) : 0u;
    }

    v4f* __restrict__ outrow = (v4f*)(out + (size_t)r * VV);

    int it = 0;
    for (int q = tid; q < Q_PER_ROW; q += 256, ++it) {
        v4f v = {0.f, 0.f, 0.f, 0.f};
        if (own_mask) {                          // rare path: this thread owns a token slot
            #pragma unroll
            for (int j = 0; j < WW; ++j) {
                if (m_it[j] == it) {
                    const float x = m_val[j];
                    const int   c = m_comp[j];
                    v.x += (c == 0) ? x : 0.f;
                    v.y += (c == 1) ? x : 0.f;
                    v.z += (c == 2) ? x : 0.f;
                    v.w += (c == 3) ? x : 0.f;
                }
            }
        }
        __builtin_nontemporal_store(v, outrow + q);   // global_store_b128, TH=NT
    }
}

// ---------------------------------------------------------------------------
extern "C" void kernel_launch(void* const* d_in, const int* in_sizes, int n_in,
                              void* d_out, int out_size, void* d_ws, size_t ws_size,
                              hipStream_t stream) {
    (void)in_sizes; (void)n_in; (void)out_size; (void)ws_size;

    const float* hidden     = (const float*)d_in[0];
    const int*   input_ids  = (const int*)d_in[1];
    const float* gate_w     = (const float*)d_in[2];
    const float* gate_b     = (const float*)d_in[3];
    const float* lag_logits = (const float*)d_in[4];
    const float* copy_scale = (const float*)d_in[5];
    float*       out        = (float*)d_out;

    float* ws_f  = (float*)d_ws;
    float* gate  = ws_f;          // ROWS floats
    float* lw    = ws_f + ROWS;   // WW floats

    // Gate GEMV: 4096 rows, 16 rows/wave, 8 waves/block -> 32 blocks.
    gate_wmma_kernel<<<ROWS / (16 * 8), 256, 0, stream>>>(
        hidden, gate_w, gate_b, lag_logits, copy_scale, gate, lw);

    // Streaming zero-fill + in-register scatter merge: one block per row.
    fill_scatter_kernel<<<ROWS, 256, 0, stream>>>(input_ids, gate, lw, out);
}